// ReyEquiv2to2_24893630447672
// MI455X (gfx1250) — compile-verified
//
#include <hip/hip_runtime.h>
#include <hip/hip_bf16.h>

typedef __attribute__((ext_vector_type(2))) float v2f;
typedef __attribute__((ext_vector_type(8))) float v8f;

#define N_  128
#define C_  16
#define B_  4
#define H_  128
#define D_  128
#define NS  (N_ * N_)

// ---------------------------------------------------------------------------
// Kernel 1: pairwise 2x2 gather -> MLP(4->128->2) via V_WMMA_F32_16X16X4_F32
// -> scatter h1 off-diagonal, accumulate h0 into diagonal (LDS), write "out".
// One block per (b,c); 16 waves per block.
// ---------------------------------------------------------------------------
__global__ void __launch_bounds__(512)
k_pair_mlp(const float* __restrict__ x,
           const float* __restrict__ W1,   // (128,4)
           const float* __restrict__ b1,   // (128)
           const float* __restrict__ W2,   // (2,128)
           const float* __restrict__ b2,   // (2)
           float* __restrict__ out)        // (B*C, N, N) workspace
{
    const int bc = blockIdx.x;                         // 0..63
    const float* __restrict__ xp = x   + (size_t)bc * NS;
    float*       __restrict__ op = out + (size_t)bc * NS;

    __shared__ float sdiag[N_];
    for (int j = threadIdx.x; j < N_; j += blockDim.x) sdiag[j] = 0.0f;
    __syncthreads();

    const int lane = threadIdx.x & 31;
    const int wave = threadIdx.x >> 5;                 // 0..15
    const int half = lane >> 4;                        // 0: lanes 0-15, 1: 16-31
    const int ln   = lane & 15;

    // tiles: t -> ii = t>>3 (0..126), jj0 = (t&7)*16
    for (int t = wave; t < (N_ - 1) * 8; t += 16) {
        const int ii  = t >> 3;
        const int jj0 = (t & 7) << 4;

        // --- A fragment: 16x4 fp32.  Row M = ln for both halves.
        // half 0 holds K0,K1 = x[p,p], x[p,q]; half 1 holds K2,K3 = x[q,p], x[q,q].
        const int p  = jj0 + ln;
        const int q  = (p + ii + 1) & (N_ - 1);
        const int r0 = half ? q : p;
        v2f a;
        a.x = xp[r0 * N_ + p];
        a.y = xp[r0 * N_ + q];

        float p0[8], p1[8];
#pragma unroll
        for (int r = 0; r < 8; ++r) { p0[r] = 0.0f; p1[r] = 0.0f; }

#pragma unroll
        for (int ht = 0; ht < 8; ++ht) {
            const int h = ht * 16 + ln;                // hidden column owned by lane
            // --- B fragment: 4x16 fp32 of W1^T. Lane column N = ln;
            // half 0 holds K0,K1; half 1 holds K2,K3.
            const int k0 = half * 2;
            v2f bf;
            bf.x = W1[h * 4 + k0];
            bf.y = W1[h * 4 + k0 + 1];

            v8f acc = {};
            acc = __builtin_amdgcn_wmma_f32_16x16x4_f32(
                      false, a, false, bf, (short)0, acc, false, false);

            const float bias = b1[h];
            const float w20  = W2[h];
            const float w21  = W2[H_ + h];
            // acc[r] = hidden[row = r + 8*half][col = h]
#pragma unroll
            for (int r = 0; r < 8; ++r) {
                float v = acc[r] + bias;
                v = v > 0.0f ? v : 0.0f;
                p0[r] += v * w20;
                p1[r] += v * w21;
            }
        }

        // reduce the 16 hidden-columns spread across each 16-lane half
#pragma unroll
        for (int r = 0; r < 8; ++r) {
#pragma unroll
            for (int m = 1; m < 16; m <<= 1) {
                p0[r] += __shfl_xor(p0[r], m, 32);
                p1[r] += __shfl_xor(p1[r], m, 32);
            }
        }

        // lanes 0 and 16 write rows 0-7 / 8-15 of this tile
        if (ln == 0) {
            const float bb0 = b2[0], bb1 = b2[1];
#pragma unroll
            for (int r = 0; r < 8; ++r) {
                const int row = r + half * 8;
                const int pp  = jj0 + row;
                const int qq  = (pp + ii + 1) & (N_ - 1);
                op[pp * N_ + qq] = p1[r] + bb1;        // off-diagonal h1
                atomicAdd(&sdiag[pp], p0[r] + bb0);    // diagonal h0 sum (ds_add_f32)
            }
        }
    }

    __syncthreads();
    const float inv = 1.0f / (float)(N_ - 1);
    for (int j = threadIdx.x; j < N_; j += blockDim.x)
        op[j * N_ + j] = sdiag[j] * inv;
}

// ---------------------------------------------------------------------------
// Kernel 2: channel mix  y[b,d,s] = relu(sum_c Wc[d,c]*out[b,c,s] + bc[d])
// GEMM (128 x 16) x (16 x 16384) per batch: 4 chained fp32 WMMAs (K=16).
// One 16x16 output tile per wave; 8 waves per block.
// ---------------------------------------------------------------------------
__global__ void __launch_bounds__(256)
k_channel_mix(const float* __restrict__ out,  // (B*C, NS)
              const float* __restrict__ Wc,   // (128,16)
              const float* __restrict__ bcv,  // (128)
              float* __restrict__ y)          // (B,128,NS)
{
    const int lane = threadIdx.x & 31;
    const int wave = threadIdx.x >> 5;
    const int half = lane >> 4;
    const int ln   = lane & 15;

    const int W  = blockIdx.x * 8 + wave;      // 0..32767
    const int b  = W >> 13;                    // 0..3
    const int dt = (W >> 10) & 7;              // 0..7
    const int st = W & 1023;                   // 0..1023
    const int d0 = dt << 4;
    const int s0 = st << 4;

    const float* __restrict__ ob = out + (size_t)b * C_ * NS;
    float*       __restrict__ yb = y   + (size_t)b * D_ * NS;

    // A fragments: Wc tile, M = d0+ln, K pairs (2*half, 2*half+1) per k-step
    v2f afr[4];
#pragma unroll
    for (int kk = 0; kk < 4; ++kk) {
        const int c = kk * 4 + half * 2;
        afr[kk].x = Wc[(d0 + ln) * C_ + c];
        afr[kk].y = Wc[(d0 + ln) * C_ + c + 1];
    }

    // B fragments: out tile, N = s0+ln, K pairs (2*half, 2*half+1) per k-step
    v2f bfr[4];
#pragma unroll
    for (int kk = 0; kk < 4; ++kk) {
        const int c = kk * 4 + half * 2;
        bfr[kk].x = ob[(size_t)c * NS + s0 + ln];
        bfr[kk].y = ob[(size_t)(c + 1) * NS + s0 + ln];
    }

    v8f acc = {};
#pragma unroll
    for (int kk = 0; kk < 4; ++kk)
        acc = __builtin_amdgcn_wmma_f32_16x16x4_f32(
                  false, afr[kk], false, bfr[kk], (short)0, acc, false, false);

    // acc[r] -> d = d0 + r + 8*half, s = s0 + ln
#pragma unroll
    for (int r = 0; r < 8; ++r) {
        const int d = d0 + r + half * 8;
        float v = acc[r] + bcv[d];
        v = v > 0.0f ? v : 0.0f;
        yb[(size_t)d * NS + s0 + ln] = v;
    }
}

// ---------------------------------------------------------------------------
extern "C" void kernel_launch(void* const* d_in, const int* in_sizes, int n_in,
                              void* d_out, int out_size, void* d_ws, size_t ws_size,
                              hipStream_t stream) {
    const float* x  = (const float*)d_in[0];
    const float* W1 = (const float*)d_in[1];
    const float* b1 = (const float*)d_in[2];
    const float* W2 = (const float*)d_in[3];
    const float* b2 = (const float*)d_in[4];
    const float* Wc = (const float*)d_in[5];
    const float* bc = (const float*)d_in[6];
    float* y   = (float*)d_out;
    float* wsO = (float*)d_ws;                 // B*C*N*N floats = 4 MB

    k_pair_mlp<<<B_ * C_, 512, 0, stream>>>(x, W1, b1, W2, b2, wsO);
    k_channel_mix<<<4096, 256, 0, stream>>>(wsO, Wc, bc, y);
}